// lstm2_841813590591
// MI455X (gfx1250) — compile-verified
//
#include <hip/hip_runtime.h>

namespace {

constexpr int Bn = 8192;
constexpr int Tn = 2048;
constexpr int Fn = 3;
constexpr int Hn = 2;
constexpr int Ln = 4;
constexpr int Gn = 4 * Hn;  // 8 gate rows

typedef float v2f __attribute__((ext_vector_type(2)));
typedef float v8f __attribute__((ext_vector_type(8)));

__device__ __forceinline__ v8f wmma_f32_16x16x4(v2f a, v2f b, v8f c) {
  // D = A(16x4) * B(4x16) + C(16x16), all fp32, one wave.
  return __builtin_amdgcn_wmma_f32_16x16x4_f32(
      /*neg_a=*/false, a, /*neg_b=*/false, b,
      /*c_mod=*/(short)0, c, /*reuse_a=*/false, /*reuse_b=*/false);
}

__device__ __forceinline__ float tanhfast(float x) {
#if __has_builtin(__builtin_amdgcn_tanhf)
  return __builtin_amdgcn_tanhf(x);   // native v_tanh_f32
#else
  float e = __expf(-2.0f * x);
  return fmaf(-2.0f, __builtin_amdgcn_rcpf(1.0f + e), 1.0f);
#endif
}

__device__ __forceinline__ float sigf(float x) {
#if __has_builtin(__builtin_amdgcn_tanhf)
  // sigmoid(x) = 0.5*tanh(x/2) + 0.5 : one TRANS op instead of exp+rcp.
  return fmaf(0.5f, __builtin_amdgcn_tanhf(0.5f * x), 0.5f);
#else
  return __builtin_amdgcn_rcpf(1.0f + __expf(-x));
#endif
}

// LSTM cell pointwise update from an 8-gate vector (i0,i1,f0,f1,g0,g1,o0,o1).
__device__ __forceinline__ void cell_update(const v8f& acc, float h[2], float c[2]) {
  const float i0 = acc[0], i1 = acc[1];
  const float f0 = acc[2], f1 = acc[3];
  const float g0 = acc[4], g1 = acc[5];
  const float o0 = acc[6], o1 = acc[7];
  c[0] = sigf(f0) * c[0] + sigf(i0) * tanhfast(g0);
  c[1] = sigf(f1) * c[1] + sigf(i1) * tanhfast(g1);
  h[0] = sigf(o0) * tanhfast(c[0]);
  h[1] = sigf(o1) * tanhfast(c[1]);
}

}  // namespace

__global__ __launch_bounds__(256) void lstm4_wmma_f32_kernel(
    const float* __restrict__ x,          // (B, T, 3)
    const float* __restrict__ w_ih0,      // (8, 3)
    const float* __restrict__ w_ih_rest,  // (3, 8, 2)
    const float* __restrict__ w_hh,       // (4, 8, 2)
    const float* __restrict__ b_ih,       // (4, 8)
    const float* __restrict__ b_hh,       // (4, 8)
    const float* __restrict__ w_out,      // (1, 2)
    const float* __restrict__ b_out,      // (1,)
    float* __restrict__ out)              // (B, T, 1)
{
  const int lane = (int)(threadIdx.x & 31u);
  const int half = lane >> 4;       // 0 -> K{0,1} slice, 1 -> K{2,3} slice
  const int m    = lane & 15;       // A-row / B-column / C-column index
  const int g    = m & 7;           // gate row (rows 8..15 duplicate 0..7)
  const int wave = (int)((blockIdx.x * blockDim.x + threadIdx.x) >> 5);
  const int bat  = wave * 16 + m;   // batch element owned by this lane

  // ---------------- A matrices (weights), resident in VGPRs ----------------
  // Layer 0, x-part. K = [x0, x1, x2, 1] so A col3 carries the bias.
  v2f A0x;
  A0x.x = (half == 0) ? w_ih0[g * Fn + 0] : w_ih0[g * Fn + 2];
  A0x.y = (half == 0) ? w_ih0[g * Fn + 1] : (b_ih[0 * Gn + g] + b_hh[0 * Gn + g]);
  // Layer 0, recurrent part. K = [h0, h1, 0, 0].
  v2f A0h;
  A0h.x = (half == 0) ? w_hh[0 * Gn * Hn + g * Hn + 0] : 0.0f;
  A0h.y = (half == 0) ? w_hh[0 * Gn * Hn + g * Hn + 1] : 0.0f;
  // Layers 1..3: A = [w_ih | w_hh], K = [h_below0, h_below1, h_rec0, h_rec1].
  v2f Al[Ln - 1];
#pragma unroll
  for (int l = 1; l < Ln; ++l) {
    Al[l - 1].x = (half == 0) ? w_ih_rest[(l - 1) * Gn * Hn + g * Hn + 0]
                              : w_hh[l * Gn * Hn + g * Hn + 0];
    Al[l - 1].y = (half == 0) ? w_ih_rest[(l - 1) * Gn * Hn + g * Hn + 1]
                              : w_hh[l * Gn * Hn + g * Hn + 1];
  }
  // Accumulator-init bias vectors for layers 1..3 (identical in every lane:
  // C vgpr r is gate r for both lane halves thanks to duplicated A rows).
  v8f biasv[Ln - 1];
#pragma unroll
  for (int l = 1; l < Ln; ++l) {
#pragma unroll
    for (int r = 0; r < Gn; ++r)
      biasv[l - 1][r] = b_ih[l * Gn + r] + b_hh[l * Gn + r];
  }

  const float wo0 = w_out[0], wo1 = w_out[1], bo = b_out[0];

  // ---------------- Recurrent state, registers only ----------------
  float h[Ln][Hn], c[Ln][Hn];
#pragma unroll
  for (int l = 0; l < Ln; ++l) {
    h[l][0] = 0.0f; h[l][1] = 0.0f;
    c[l][0] = 0.0f; c[l][1] = 0.0f;
  }

  const float* xrow = x + (size_t)bat * Tn * Fn;
  float* orow = out + (size_t)bat * Tn;
  const v8f zero = {};

  for (int t = 0; t < Tn; ++t) {
    const float x0 = xrow[0], x1 = xrow[1], x2 = xrow[2];
    xrow += Fn;

    // ---- Layer 0: two accumulating fp32 WMMAs (x+bias, then recurrent) ----
    v2f Bx;
    Bx.x = (half == 0) ? x0 : x2;
    Bx.y = (half == 0) ? x1 : 1.0f;   // K=3 row of ones multiplies the bias col
    v8f acc = wmma_f32_16x16x4(A0x, Bx, zero);
    v2f Bh;
    Bh.x = (half == 0) ? h[0][0] : 0.0f;
    Bh.y = (half == 0) ? h[0][1] : 0.0f;
    acc = wmma_f32_16x16x4(A0h, Bh, acc);
    cell_update(acc, h[0], c[0]);

    // ---- Layers 1..3: single WMMA each, bias preloaded in accumulator ----
#pragma unroll
    for (int l = 1; l < Ln; ++l) {
      v2f Bl;
      Bl.x = (half == 0) ? h[l - 1][0] : h[l][0];
      Bl.y = (half == 0) ? h[l - 1][1] : h[l][1];
      v8f a = wmma_f32_16x16x4(Al[l - 1], Bl, biasv[l - 1]);
      cell_update(a, h[l], c[l]);
    }

    // ---- Output projection; lanes 16..31 are duplicates, don't store ----
    if (half == 0)
      orow[t] = fmaf(h[Ln - 1][0], wo0, fmaf(h[Ln - 1][1], wo1, bo));
  }
}

extern "C" void kernel_launch(void* const* d_in, const int* in_sizes, int n_in,
                              void* d_out, int out_size, void* d_ws, size_t ws_size,
                              hipStream_t stream) {
  (void)in_sizes; (void)n_in; (void)d_ws; (void)ws_size; (void)out_size;
  const float* x         = (const float*)d_in[0];
  const float* w_ih0     = (const float*)d_in[1];
  const float* w_ih_rest = (const float*)d_in[2];
  const float* w_hh      = (const float*)d_in[3];
  const float* b_ih      = (const float*)d_in[4];
  const float* b_hh      = (const float*)d_in[5];
  const float* w_out     = (const float*)d_in[6];
  const float* b_out     = (const float*)d_in[7];
  float* out             = (float*)d_out;

  // 16 batch elements per wave -> 512 waves; 8 waves (256 thr) per block.
  const int waves  = Bn / 16;          // 512
  const int blocks = waves / 8;        // 64
  lstm4_wmma_f32_kernel<<<blocks, 256, 0, stream>>>(
      x, w_ih0, w_ih_rest, w_hh, b_ih, b_hh, w_out, b_out, out);
}